// PlenoxelGrid_44770739093770
// MI455X (gfx1250) — compile-verified
//
#include <hip/hip_runtime.h>

#define GRID_N 256

typedef float v4f __attribute__((ext_vector_type(4)));

__device__ __forceinline__ v4f lerp4(const v4f a, const v4f b, const float t) {
    // matches reference: a*(1-t) + b*t (scalar t broadcasts across the vector)
    return a * (1.0f - t) + b * t;
}

__global__ __launch_bounds__(256) void plenoxel_trilerp_kernel(
    const float* __restrict__ points,      // [N,3] fp32, streamed once (NT)
    const v4f* __restrict__ grid,          // [256,256,256] of float4, hot in L2 (RT)
    v4f* __restrict__ out,                 // [N] float4, streamed once (NT)
    int n)
{
    const int i = blockIdx.x * blockDim.x + threadIdx.x;
    if (i >= n) return;

    // --- streaming point load (non-temporal: don't pollute L2, keep it for the grid)
    const float* pp = points + 3ull * (unsigned)i;
    const float px = __builtin_nontemporal_load(pp + 0);
    const float py = __builtin_nontemporal_load(pp + 1);
    const float pz = __builtin_nontemporal_load(pp + 2);

    // normalized = (p - (-1)) / 2 ; grid_coords = normalized * 255
    // (p+1)*0.5f*255 == (p+1)*127.5f bit-exactly (mul by 0.5 is exact)
    const float gx = (px + 1.0f) * 127.5f;
    const float gy = (py + 1.0f) * 127.5f;
    const float gz = (pz + 1.0f) * 127.5f;

    // frac computed BEFORE clamp, matching reference
    const float fx = floorf(gx), fy = floorf(gy), fz = floorf(gz);
    const float dx = gx - fx,    dy = gy - fy,    dz = gz - fz;

    const int x0 = min(max((int)fx, 0), GRID_N - 2);
    const int y0 = min(max((int)fy, 0), GRID_N - 2);
    const int z0 = min(max((int)fz, 0), GRID_N - 2);

    // flat index in float4 units: ((x*256)+y)*256 + z
    const unsigned sy   = GRID_N;            // +1 in y  => +256 float4
    const unsigned sx   = GRID_N * GRID_N;   // +1 in x  => +65536 float4
    const unsigned base = ((unsigned)x0 * GRID_N + (unsigned)y0) * GRID_N + (unsigned)z0;

    // 8 corner gathers: z-adjacent pairs are contiguous 32B => 4 hot regions.
    // Issue all 8 b128 loads before use for maximum memory-level parallelism.
    const v4f c000 = grid[base];
    const v4f c001 = grid[base + 1];
    const v4f c010 = grid[base + sy];
    const v4f c011 = grid[base + sy + 1];
    const v4f c100 = grid[base + sx];
    const v4f c101 = grid[base + sx + 1];
    const v4f c110 = grid[base + sx + sy];
    const v4f c111 = grid[base + sx + sy + 1];

    // trilinear, reference order: x then y then z
    const v4f c00 = lerp4(c000, c100, dx);
    const v4f c01 = lerp4(c001, c101, dx);
    const v4f c10 = lerp4(c010, c110, dx);
    const v4f c11 = lerp4(c011, c111, dx);

    const v4f c0 = lerp4(c00, c10, dy);
    const v4f c1 = lerp4(c01, c11, dy);

    const v4f c = lerp4(c0, c1, dz);

    // streaming store (non-temporal): output is never re-read
    __builtin_nontemporal_store(c, out + (unsigned)i);
}

extern "C" void kernel_launch(void* const* d_in, const int* in_sizes, int n_in,
                              void* d_out, int out_size, void* d_ws, size_t ws_size,
                              hipStream_t stream) {
    const float* points = (const float*)d_in[0];         // [N,3] fp32
    const v4f*   grid   = (const v4f*)d_in[1];           // [256,256,256,4] fp32
    v4f*         out    = (v4f*)d_out;                   // [N,4] fp32

    const int n = in_sizes[0] / 3;                       // 4,194,304 points
    const int block = 256;                               // 8 wave32 per block
    const int nblk = (n + block - 1) / block;            // 16384 blocks

    plenoxel_trilerp_kernel<<<nblk, block, 0, stream>>>(points, grid, out, n);
}